// ScaledGenerator_75866302316776
// MI455X (gfx1250) — compile-verified
//
#include <hip/hip_runtime.h>

#define B_  64
#define S_  128
#define D_  64
#define NB_ 8
#define TPB 512   // 16 waves (wave32): one 16x16 output tile per wave

typedef float v2f __attribute__((ext_vector_type(2)));
typedef float v8f __attribute__((ext_vector_type(8)));

// Wave-cooperative 64x64x64 fp32 matmul: Dst = (TRANS_A ? A^T : A) @ B, in LDS.
// V_WMMA_F32_16X16X4_F32; layouts per CDNA5 ISA 7.12.2.
template<bool TRANS_A>
__device__ inline void mm64(float* Dst, const float* Am, const float* Bm) {
    __syncthreads();
    const int wave = threadIdx.x >> 5;
    const int lane = threadIdx.x & 31;
    const int I    = wave >> 2;
    const int J    = wave & 3;
    const int half = lane >> 4;
    const int l16  = lane & 15;
    const int row  = 16 * I + l16;
    const int col  = 16 * J + l16;
    v8f c = (v8f)0.0f;
#pragma unroll
    for (int kb = 0; kb < 64; kb += 4) {
        const int kk = kb + 2 * half;
        v2f a, b;
        if (TRANS_A) {
            a.x = Am[(kk + 0) * 64 + row];
            a.y = Am[(kk + 1) * 64 + row];
        } else {
            a.x = Am[row * 64 + kk + 0];
            a.y = Am[row * 64 + kk + 1];
        }
        b.x = Bm[(kk + 0) * 64 + col];
        b.y = Bm[(kk + 1) * 64 + col];
        c = __builtin_amdgcn_wmma_f32_16x16x4_f32(false, a, false, b,
                                                  (short)0, c, false, false);
    }
    const int orow = 16 * I + 8 * half;
    const int ocol = 16 * J + l16;
#pragma unroll
    for (int v = 0; v < 8; ++v)
        Dst[(orow + v) * 64 + ocol] = c[v];
    __syncthreads();
}

// Horner-step matmul with register-resident A fragments (X is loop-invariant):
//   Dst = postScale * (A @ B) + (addI ? I : 0), fused into the fragment store.
__device__ inline void mm64_regA(float* Dst, const v2f* __restrict__ af,
                                 const float* Bm, float postScale, bool addI,
                                 int I, int J, int half, int l16) {
    __syncthreads();
    const int col = 16 * J + l16;
    v8f c = (v8f)0.0f;
#pragma unroll
    for (int kb16 = 0; kb16 < 16; ++kb16) {
        const int kk = 4 * kb16 + 2 * half;
        v2f b;
        b.x = Bm[(kk + 0) * 64 + col];
        b.y = Bm[(kk + 1) * 64 + col];
        c = __builtin_amdgcn_wmma_f32_16x16x4_f32(false, af[kb16], false, b,
                                                  (short)0, c, false, false);
    }
    const int orow = 16 * I + 8 * half;
#pragma unroll
    for (int v = 0; v < 8; ++v) {
        float val = c[v] * postScale;
        if (addI && (orow + v) == col) val += 1.0f;
        Dst[(orow + v) * 64 + col] = val;
    }
    __syncthreads();
}

// Kernel 1: per-batch generator  A = alpha*(K - K^T) - beta * R^T R, plus inf-norm.
__global__ void __launch_bounds__(TPB)
buildA(const float* __restrict__ kc_g, const float* __restrict__ rc_g,
       const float* __restrict__ alpha_g, const float* __restrict__ beta_g,
       const float* __restrict__ Kb, const float* __restrict__ Rb,
       float* __restrict__ gA, float* __restrict__ gNorm) {
    __shared__ float Ks[4096];
    __shared__ float Rs[4096];
    __shared__ float Mb[4096];
    __shared__ float rowsum[64];
    const int b   = blockIdx.x;
    const int tid = threadIdx.x;

    float kc[NB_], rc[NB_];
#pragma unroll
    for (int k = 0; k < NB_; ++k) {
        kc[k] = kc_g[b * NB_ + k];
        rc[k] = rc_g[b * NB_ + k];
    }
#pragma unroll
    for (int e = 0; e < 8; ++e) {
        const int idx = tid + e * TPB;
        float sk = 0.f, sr = 0.f;
#pragma unroll
        for (int k = 0; k < NB_; ++k) {
            sk += kc[k] * Kb[k * 4096 + idx];
            sr += rc[k] * Rb[k * 4096 + idx];
        }
        Ks[idx] = sk;
        Rs[idx] = sr;
    }

    // Mb = Rs^T @ Rs  (fp32 WMMA)
    mm64<true>(Mb, Rs, Rs);

    const float al = alpha_g[b], be = beta_g[b];
#pragma unroll
    for (int e = 0; e < 8; ++e) {
        const int idx = tid + e * TPB;
        const int i = idx >> 6, j = idx & 63;
        const float a = al * (Ks[i * 64 + j] - Ks[j * 64 + i]) - be * Mb[idx];
        gA[b * 4096 + idx] = a;
        Rs[idx] = a;   // Rs now free; reuse for norm
    }
    __syncthreads();
    if (tid < 64) {
        float s = 0.f;
        for (int j = 0; j < 64; ++j) s += fabsf(Rs[tid * 64 + j]);
        rowsum[tid] = s;
    }
    __syncthreads();
    if (tid == 0) {
        float m = 0.f;
        for (int i = 0; i < 64; ++i) m = fmaxf(m, rowsum[i]);
        gNorm[b] = m;
    }
}

// Kernel 2: per (b,s): P = expm(t*A) via scaling-and-squaring + order-8 Taylor
// (Horner, A fragments register-resident, scale+identity fused into WMMA store);
// fused out = P @ z0.
__global__ void __launch_bounds__(TPB)
expmProp(const float* __restrict__ gA, const float* __restrict__ gNorm,
         const float* __restrict__ z0, const float* __restrict__ ts,
         float* __restrict__ out) {
    __shared__ float sX[4096];
    __shared__ float bufA[4096];
    __shared__ float bufB[4096];
    __shared__ float zv[64];
    const int bs  = blockIdx.x;
    const int b   = bs >> 7;          // / S_
    const int si  = bs & (S_ - 1);
    const int tid = threadIdx.x;

    const int wave = tid >> 5;
    const int lane = tid & 31;
    const int I    = wave >> 2;
    const int J    = wave & 3;
    const int half = lane >> 4;
    const int l16  = lane & 15;
    const int arow = 16 * I + l16;    // A-fragment row for this lane

    const float t = ts[si];
    float nrm = gNorm[b] * fabsf(t);
    int sq = 0;
    while (nrm > 0.5f && sq < 30) { nrm *= 0.5f; ++sq; }          // theta = 0.5
    const float scale = t * exp2f((float)(-sq));

    float* T = bufA;
    float* W = bufB;
    // X = scale*A ; T = I + X/8 (innermost Horner term)
#pragma unroll
    for (int e = 0; e < 8; ++e) {
        const int idx = tid + e * TPB;
        const float x = gA[b * 4096 + idx] * scale;
        sX[idx] = x;
        const int i = idx >> 6, j = idx & 63;
        T[idx] = x * 0.125f + ((i == j) ? 1.0f : 0.0f);
    }
    __syncthreads();

    // Preload this wave's A fragments of X once (invariant over Horner loop):
    // 16 x v2f = 32 VGPRs, kills the A-side LDS traffic for 7 matmuls.
    v2f af[16];
#pragma unroll
    for (int kb16 = 0; kb16 < 16; ++kb16) {
        const int kk = 4 * kb16 + 2 * half;
        af[kb16].x = sX[arow * 64 + kk + 0];
        af[kb16].y = sX[arow * 64 + kk + 1];
    }

    // Horner: for k=7..1:  T <- I + (X @ T)/k   (scale + identity fused in store)
    for (int k = 7; k >= 1; --k) {
        mm64_regA(W, af, T, 1.0f / (float)k, true, I, J, half, l16);
        float* tmp = T; T = W; W = tmp;
    }
    // Repeated squaring: T <- T @ T, sq times (operands change: LDS version)
    for (int r = 0; r < sq; ++r) {
        mm64<false>(W, T, T);
        float* tmp = T; T = W; W = tmp;
    }
    __syncthreads();
    if (tid < 64) zv[tid] = z0[b * 64 + tid];
    __syncthreads();
    // z_evolved[b,s,o] = sum_i P[o,i] * z0[i]; complex64 output (imag = 0)
    if (tid < 64) {
        float acc = 0.f;
        for (int j = 0; j < 64; ++j) acc += T[tid * 64 + j] * zv[j];
        const int cidx = bs * 64 + tid;          // (b*S + s)*D + o
        out[2 * cidx + 0] = acc;
        out[2 * cidx + 1] = 0.0f;
    }
}

extern "C" void kernel_launch(void* const* d_in, const int* in_sizes, int n_in,
                              void* d_out, int out_size, void* d_ws, size_t ws_size,
                              hipStream_t stream) {
    const float* z0    = (const float*)d_in[0];
    const float* ts    = (const float*)d_in[1];
    const float* alpha = (const float*)d_in[2];
    const float* beta  = (const float*)d_in[3];
    const float* kc    = (const float*)d_in[4];
    const float* rc    = (const float*)d_in[5];
    const float* Kb    = (const float*)d_in[6];
    const float* Rb    = (const float*)d_in[7];

    float* gA    = (float*)d_ws;            // B*64*64 floats = 1 MB
    float* gNorm = gA + B_ * D_ * D_;       // B floats

    buildA<<<B_, TPB, 0, stream>>>(kc, rc, alpha, beta, Kb, Rb, gA, gNorm);
    expmProp<<<B_ * S_, TPB, 0, stream>>>(gA, gNorm, z0, ts, (float*)d_out);
}